// DiagonalLinear_37211596652828
// MI455X (gfx1250) — compile-verified
//
#include <hip/hip_runtime.h>
#include <stdint.h>

// out[r, c] = x[r, c] * w[c] + b[c]   (8192 x 4096, fp32) — HBM-bandwidth bound.
// Strategy: per-block column tile of 1024 floats; stage w/b tile to LDS via the
// CDNA5 async global->LDS path, hoist to registers, then stream rows with
// non-temporal b128 loads/stores.

typedef float v4f __attribute__((ext_vector_type(4)));
typedef int   v4i __attribute__((ext_vector_type(4)));

#define GLOBAL_AS __attribute__((address_space(1)))
#define LDS_AS    __attribute__((address_space(3)))

#define TPB 256
#define COLS_PER_BLOCK (TPB * 4)   // 1024 floats per block column-tile

__global__ __launch_bounds__(TPB) void diag_linear_vec4(
    const float* __restrict__ x,
    const float* __restrict__ w,
    const float* __restrict__ bias,
    float* __restrict__ out,
    int n_rows, int d)
{
    __shared__ float s_w[COLS_PER_BLOCK];
    __shared__ float s_b[COLS_PER_BLOCK];

    const int t   = threadIdx.x;
    const int col = blockIdx.x * COLS_PER_BLOCK + t * 4;

#if __has_builtin(__builtin_amdgcn_global_load_async_to_lds_b128)
    // CDNA5 async copy path: one b128 per thread for weight and bias tiles.
    // Builtin signature (per hipcc diagnostic): (v4i AS1*, v4i AS3*, Imm off, Imm cpol)
    __builtin_amdgcn_global_load_async_to_lds_b128(
        (GLOBAL_AS v4i*)(w + col),
        (LDS_AS    v4i*)(s_w + t * 4),
        /*offset=*/0, /*cpol=*/0);
    __builtin_amdgcn_global_load_async_to_lds_b128(
        (GLOBAL_AS v4i*)(bias + col),
        (LDS_AS    v4i*)(s_b + t * 4),
        /*offset=*/0, /*cpol=*/0);
    // Each wave drains its own ASYNCcnt before the barrier; after the barrier
    // every wave's async LDS writes are therefore visible block-wide.
    asm volatile("s_wait_asynccnt 0x0" ::: "memory");
#else
    *(v4f*)(s_w + t * 4) = *(const v4f*)(w + col);
    *(v4f*)(s_b + t * 4) = *(const v4f*)(bias + col);
#endif
    __syncthreads();

    // Hoist this thread's fixed column coefficients into registers.
    const v4f wv = *(const v4f*)(s_w + t * 4);
    const v4f bv = *(const v4f*)(s_b + t * 4);

    // Grid-stride over rows; x/out are touched exactly once -> non-temporal.
    for (int row = blockIdx.y; row < n_rows; row += gridDim.y) {
        const float* px = x   + (size_t)row * d + col;
        float*       po = out + (size_t)row * d + col;
        v4f xv = __builtin_nontemporal_load((const v4f*)px);
        v4f r  = xv * wv + bv;   // contracts to packed FMA
        __builtin_nontemporal_store(r, (v4f*)po);
    }
}

// Generic fallback (d not a multiple of the tile width) — not used for D=4096.
__global__ __launch_bounds__(TPB) void diag_linear_scalar(
    const float* __restrict__ x,
    const float* __restrict__ w,
    const float* __restrict__ bias,
    float* __restrict__ out,
    long total, int d)
{
    long i      = (long)blockIdx.x * blockDim.x + threadIdx.x;
    long stride = (long)gridDim.x * blockDim.x;
    for (; i < total; i += stride) {
        int c = (int)(i % d);
        out[i] = fmaf(x[i], w[c], bias[c]);
    }
}

extern "C" void kernel_launch(void* const* d_in, const int* in_sizes, int n_in,
                              void* d_out, int out_size, void* d_ws, size_t ws_size,
                              hipStream_t stream) {
    const float* x = (const float*)d_in[0];
    const float* w = (const float*)d_in[1];
    const float* b = (const float*)d_in[2];
    float* out     = (float*)d_out;

    const int total = in_sizes[0];   // N_TOKENS * D
    const int d     = in_sizes[1];   // D (weight length)

    if (d % COLS_PER_BLOCK == 0) {
        const int n_rows = total / d;
        dim3 grid(d / COLS_PER_BLOCK, n_rows < 1024 ? n_rows : 1024);
        diag_linear_vec4<<<grid, TPB, 0, stream>>>(x, w, b, out, n_rows, d);
    } else {
        diag_linear_scalar<<<2048, TPB, 0, stream>>>(x, w, b, out, (long)total, d);
    }
}